// NTM_82798379532350
// MI455X (gfx1250) — compile-verified
//
#include <hip/hip_runtime.h>
#include <math.h>

// ---------------------------------------------------------------------------
// NTM single step for MI455X (gfx1250, wave32).
// Bandwidth-bound: ~575 MB fp32 weights/step -> ~25us floor at 23.3 TB/s.
// The dominant we_W stream (537 MB) runs through V_WMMA_F32_16X16X4_F32.
// ---------------------------------------------------------------------------

typedef __attribute__((ext_vector_type(2))) float v2f;
typedef __attribute__((ext_vector_type(8))) float v8f;

#define V_  64
#define H_  1024
#define N_  2048
#define M_  64
#define IN_ 129
#define NM_ (N_ * M_)

// d_out layout (floats)
#define OUT_OUT   0
#define OUT_READ  64
#define OUT_WR    128
#define OUT_WW    2176
#define OUT_H     4224
#define OUT_C     5248
#define OUT_MEM   6272

// workspace layout (floats)
#define WS_GATES  0        // 4096
#define WS_CO     4096     // 1024
#define WS_K      6144     // 128  (read head k @ +0, write head k @ +64)
#define WS_A      6272     // 64
#define WS_SCAL   6336     // 8:  [h*4 + {0:beta,1:g,2:gamma,3:kn}]
#define WS_SLOG   6344     // 4096
#define WS_SIM    10440    // 4096
#define WS_W      14536    // 4096 (read @ +0, write @ +2048)
#define WS_ERASE  18632    // 64
#define WS_RED    18696    // 64

struct InPtrs {
  const float *x, *prev_read, *prev_read_w, *prev_write_w, *h, *c, *memory;
  const float *W_ih, *W_hh, *b_ih, *b_hh;
  const float *rk_W, *rk_b, *rbeta_W, *rbeta_b, *rg_W, *rg_b, *rs_W, *rs_b,
              *rgamma_W, *rgamma_b;
  const float *wk_W, *wk_b, *wbeta_W, *wbeta_b, *wg_W, *wg_b, *ws_W, *ws_b,
              *wgamma_W, *wgamma_b;
  const float *we_W, *we_b, *wa_W, *wa_b, *fc_W, *fc_b;
};

__device__ __forceinline__ float sigmoidf_(float x) {
  return 1.0f / (1.0f + expf(-x));
}
__device__ __forceinline__ float softplusf_(float x) {
  return (x > 20.0f) ? x : log1pf(expf(x));
}
__device__ __forceinline__ float wave_reduce_sum(float v) {
  #pragma unroll
  for (int o = 16; o > 0; o >>= 1) v += __shfl_down(v, o, 32);
  return v;  // valid in lane 0
}
// dot of a 1024-float row against a 1024-float vector, one wave, float4 loads
__device__ __forceinline__ float wave_dot1024(const float* __restrict__ w,
                                              const float* __restrict__ v) {
  const int lane = threadIdx.x & 31;
  float s = 0.0f;
  for (int j = lane * 4; j < 1024; j += 128) {
    const float4 a = *(const float4*)(w + j);
    const float4 b = *(const float4*)(v + j);
    s = fmaf(a.x, b.x, s); s = fmaf(a.y, b.y, s);
    s = fmaf(a.z, b.z, s); s = fmaf(a.w, b.w, s);
  }
  return wave_reduce_sum(s);
}

// ---------------- K1: LSTM gates (4096 rows, 1 wave/row) -------------------
__global__ void k_lstm_gates(InPtrs P, float* __restrict__ gates) {
  const int row  = blockIdx.x * 8 + (threadIdx.x >> 5);
  const int lane = threadIdx.x & 31;
  float s = 0.0f;
  const float* wr = P.W_ih + (size_t)row * IN_;
  for (int j = lane; j < IN_; j += 32) {
    const float ci = (j < V_ + 1) ? P.x[j] : P.prev_read[j - (V_ + 1)];
    s = fmaf(wr[j], ci, s);
  }
  const float* whr = P.W_hh + (size_t)row * H_;
  for (int j = lane * 4; j < H_; j += 128) {
    const float4 a = *(const float4*)(whr + j);
    const float4 b = *(const float4*)(P.h + j);
    s = fmaf(a.x, b.x, s); s = fmaf(a.y, b.y, s);
    s = fmaf(a.z, b.z, s); s = fmaf(a.w, b.w, s);
  }
  s = wave_reduce_sum(s);
  if (lane == 0) gates[row] = s + P.b_ih[row] + P.b_hh[row];
}

// ---------------- K2: LSTM cell elementwise + zero erase accum -------------
__global__ void k_lstm_cell(InPtrs P, const float* __restrict__ gates,
                            float* __restrict__ co, float* __restrict__ dout,
                            float* __restrict__ erase_acc) {
  const int t = blockIdx.x * blockDim.x + threadIdx.x;
  if (t < H_) {
    const float ig = gates[t], fg = gates[H_ + t], gg = gates[2 * H_ + t],
                og = gates[3 * H_ + t];
    const float cn = sigmoidf_(fg) * P.c[t] + sigmoidf_(ig) * tanhf(gg);
    const float hn = sigmoidf_(og) * tanhf(cn);
    co[t] = hn;
    dout[OUT_H + t] = hn;
    dout[OUT_C + t] = cn;
  }
  if (blockIdx.x == 0 && threadIdx.x < M_) erase_acc[threadIdx.x] = 0.0f;
}

// ---------------- K3: small matvecs (k, a, beta/g/gamma) -------------------
__global__ void k_small(InPtrs P, const float* __restrict__ co,
                        float* __restrict__ kv, float* __restrict__ av,
                        float* __restrict__ scal) {
  const int W = blockIdx.x * 8 + (threadIdx.x >> 5);
  const int lane = threadIdx.x & 31;
  if (W >= 198) return;
  float r;
  if (W < 64) {
    r = wave_dot1024(P.rk_W + (size_t)W * H_, co);
    if (lane == 0) kv[W] = r + P.rk_b[W];
  } else if (W < 128) {
    const int q = W - 64;
    r = wave_dot1024(P.wk_W + (size_t)q * H_, co);
    if (lane == 0) kv[64 + q] = r + P.wk_b[q];
  } else if (W < 192) {
    const int q = W - 128;
    r = wave_dot1024(P.wa_W + (size_t)q * H_, co);
    if (lane == 0) av[q] = r + P.wa_b[q];
  } else {
    const int q = W - 192;  // 0..5
    const float* Wm; const float* bm;
    switch (q) {
      case 0: Wm = P.rbeta_W;  bm = P.rbeta_b;  break;
      case 1: Wm = P.rg_W;     bm = P.rg_b;     break;
      case 2: Wm = P.rgamma_W; bm = P.rgamma_b; break;
      case 3: Wm = P.wbeta_W;  bm = P.wbeta_b;  break;
      case 4: Wm = P.wg_W;     bm = P.wg_b;     break;
      default: Wm = P.wgamma_W; bm = P.wgamma_b; break;
    }
    r = wave_dot1024(Wm, co) ;
    if (lane == 0) {
      r += bm[0];
      const int slot = (q < 3) ? q : (4 + (q - 3));
      float val;
      const int kind = (q < 3) ? q : (q - 3);
      if (kind == 0) val = softplusf_(r);
      else if (kind == 1) val = sigmoidf_(r);
      else val = 1.0f + softplusf_(r);
      scal[slot] = val;
    }
  }
}

// ---------------- K3b: key norms -------------------------------------------
__global__ void k_knorm(const float* __restrict__ kv, float* __restrict__ scal) {
  const int h = blockIdx.x;
  const int lane = threadIdx.x;
  const float a = kv[h * 64 + lane];
  const float b = kv[h * 64 + 32 + lane];
  float s = wave_reduce_sum(a * a + b * b);
  if (lane == 0) scal[h * 4 + 3] = fmaxf(sqrtf(s), 1e-8f);
}

// ---------------- K4: shift logits s = co @ sW.T + sb ----------------------
__global__ void k_slog(InPtrs P, const float* __restrict__ co,
                       float* __restrict__ slog) {
  const int W = blockIdx.x * 8 + (threadIdx.x >> 5);  // 0..4095
  const int hgt = W >> 11, n = W & (N_ - 1);
  const float* sW = hgt ? P.ws_W : P.rs_W;
  const float* sb = hgt ? P.ws_b : P.rs_b;
  const float r = wave_dot1024(sW + (size_t)n * H_, co);
  if ((threadIdx.x & 31) == 0) slog[hgt * N_ + n] = r + sb[n];
}

// ---------------- K5: cosine similarity ------------------------------------
__global__ void k_sim(InPtrs P, const float* __restrict__ kv,
                      const float* __restrict__ scal, float* __restrict__ sim) {
  const int t = blockIdx.x * blockDim.x + threadIdx.x;  // 0..4095
  const int hgt = t >> 11, n = t & (N_ - 1);
  const float* mrow = P.memory + (size_t)n * M_;
  const float* kk = kv + hgt * 64;
  float dp = 0.0f, nrm = 0.0f;
  for (int m = 0; m < M_; m += 4) {
    const float4 mv = *(const float4*)(mrow + m);
    const float4 kw = *(const float4*)(kk + m);
    dp  = fmaf(mv.x, kw.x, dp);  dp = fmaf(mv.y, kw.y, dp);
    dp  = fmaf(mv.z, kw.z, dp);  dp = fmaf(mv.w, kw.w, dp);
    nrm = fmaf(mv.x, mv.x, nrm); nrm = fmaf(mv.y, mv.y, nrm);
    nrm = fmaf(mv.z, mv.z, nrm); nrm = fmaf(mv.w, mv.w, nrm);
  }
  const float mn = fmaxf(sqrtf(nrm), 1e-8f);
  sim[hgt * N_ + n] = dp / (mn * scal[hgt * 4 + 3]);
}

// ---------------- K6: full addressing pipeline per head --------------------
// blockDim = 1024, gridDim = 2 (head). Softmax(s), softmax(beta*sim),
// interpolate, circular convolution, sharpen, normalize.
__device__ __forceinline__ float blk_reduce(float* red, float v, int t, bool ismax) {
  red[t] = v;
  __syncthreads();
  for (int s = 512; s > 0; s >>= 1) {
    if (t < s) red[t] = ismax ? fmaxf(red[t], red[t + s]) : (red[t] + red[t + s]);
    __syncthreads();
  }
  const float r = red[0];
  __syncthreads();
  return r;
}

__global__ __launch_bounds__(1024)
void k_address(InPtrs P, const float* __restrict__ slog,
               const float* __restrict__ sim, const float* __restrict__ scal,
               float* __restrict__ wout, float* __restrict__ dout) {
  __shared__ float s_sm[N_];
  __shared__ float wg[N_];
  __shared__ float red[1024];
  const int h = blockIdx.x;
  const int t = threadIdx.x;
  const float beta  = scal[h * 4 + 0];
  const float g     = scal[h * 4 + 1];
  const float gamma = scal[h * 4 + 2];

  // softmax of shift logits -> s_sm
  const float l0 = slog[h * N_ + t], l1 = slog[h * N_ + 1024 + t];
  const float mx = blk_reduce(red, fmaxf(l0, l1), t, true);
  const float e0 = expf(l0 - mx), e1 = expf(l1 - mx);
  const float ssum = blk_reduce(red, e0 + e1, t, false);
  s_sm[t] = e0 / ssum;
  s_sm[1024 + t] = e1 / ssum;

  // softmax of beta*sim -> w_c ; interpolate with prev_read_w -> wg
  const float b0 = beta * sim[h * N_ + t], b1 = beta * sim[h * N_ + 1024 + t];
  const float mx2 = blk_reduce(red, fmaxf(b0, b1), t, true);
  const float f0 = expf(b0 - mx2), f1 = expf(b1 - mx2);
  const float csum = blk_reduce(red, f0 + f1, t, false);
  wg[t]        = g * (f0 / csum) + (1.0f - g) * P.prev_read_w[t];
  wg[1024 + t] = g * (f1 / csum) + (1.0f - g) * P.prev_read_w[1024 + t];
  __syncthreads();

  // circular convolution: w_t[i] = sum_j wg[j] * s[(i-j) mod N]
  const int i0 = t, i1 = t + 1024;
  float a0 = 0.0f, a1 = 0.0f;
  for (int j = 0; j < N_; ++j) {
    const float w = wg[j];  // broadcast read
    a0 = fmaf(w, s_sm[(i0 - j) & (N_ - 1)], a0);
    a1 = fmaf(w, s_sm[(i1 - j) & (N_ - 1)], a1);
  }
  // sharpen + normalize
  const float p0 = powf(fmaxf(a0, 0.0f), gamma);
  const float p1 = powf(fmaxf(a1, 0.0f), gamma);
  const float tot = blk_reduce(red, p0 + p1, t, false) + 1e-16f;
  float* dd = dout + (h ? OUT_WW : OUT_WR);
  const float w0 = p0 / tot, w1 = p1 / tot;
  wout[h * N_ + t] = w0;        wout[h * N_ + 1024 + t] = w1;
  dd[t] = w0;                   dd[1024 + t] = w1;
}

// ---------------- K7: read_out = w_r @ memory ------------------------------
__global__ void k_readout(InPtrs P, const float* __restrict__ wr,
                          float* __restrict__ red_ws, float* __restrict__ dout) {
  __shared__ float lds[256];
  const int t = threadIdx.x;
  const int m = t & 63, slice = t >> 6;
  float acc = 0.0f;
  for (int n = slice; n < N_; n += 4)
    acc = fmaf(wr[n], P.memory[(size_t)n * M_ + m], acc);
  lds[t] = acc;
  __syncthreads();
  if (slice == 0) {
    const float r = lds[t] + lds[t + 64] + lds[t + 128] + lds[t + 192];
    red_ws[m] = r;
    dout[OUT_READ + m] = r;
  }
}

// ---------------- K8: fc output --------------------------------------------
__global__ void k_fc(InPtrs P, const float* __restrict__ co,
                     const float* __restrict__ red_ws, float* __restrict__ dout) {
  const int v = blockIdx.x * 8 + (threadIdx.x >> 5);  // 0..63
  const int lane = threadIdx.x & 31;
  const float* row = P.fc_W + (size_t)v * (H_ + M_);
  float s = 0.0f;
  for (int j = lane * 4; j < H_; j += 128) {
    const float4 a = *(const float4*)(row + j);
    const float4 b = *(const float4*)(co + j);
    s = fmaf(a.x, b.x, s); s = fmaf(a.y, b.y, s);
    s = fmaf(a.z, b.z, s); s = fmaf(a.w, b.w, s);
  }
  for (int j = lane; j < M_; j += 32) s = fmaf(row[H_ + j], red_ws[j], s);
  s = wave_reduce_sum(s);
  if (lane == 0) dout[OUT_OUT + v] = sigmoidf_(s + P.fc_b[v]);
}

// ---------------- K9: BIG erase-gate stream via WMMA -----------------------
// erase[m] = sum_n w_w[n] * sigmoid(we_W[n*64+m,:] . co + we_b[n*64+m])
// One wave owns 16 consecutive rows; A fragments follow the ISA 16x4 f32
// layout (lanes 0-15: K+0,K+1 / lanes 16-31: K+2,K+3), so the two half-waves
// together stream each row contiguously. B broadcasts co[k..k+3] over all 16
// columns -> D column 0 is the matvec; lanes 0/16 hold the 16 row results.
__global__ __launch_bounds__(256)
void k_erase_wmma(InPtrs P, const float* __restrict__ co,
                  const float* __restrict__ w_write,
                  float* __restrict__ erase_acc) {
  __shared__ float co_l[H_];
  __shared__ float er[M_];
  const int t = threadIdx.x;
  for (int j = t; j < H_; j += 256) co_l[j] = co[j];
  if (t < M_) er[t] = 0.0f;
  __syncthreads();

  const int wave = blockIdx.x * 8 + (t >> 5);   // 0..8191
  const int lane = t & 31;
  const int r0 = wave * 16;                     // 16 rows per wave
  const int n = r0 >> 6;                        // memory slot (rows share n)
  const int mbase = r0 & 63;
  const int row = r0 + (lane & 15);
  const int koff = (lane >> 4) << 1;            // 0 for lanes<16, 2 otherwise
  const float* wrow = P.we_W + (size_t)row * H_;

  v8f acc0 = {0.f,0.f,0.f,0.f,0.f,0.f,0.f,0.f};
  v8f acc1 = {0.f,0.f,0.f,0.f,0.f,0.f,0.f,0.f};
  for (int k0 = 0; k0 < H_; k0 += 8) {
    v2f a0 = *(const v2f*)(wrow + k0 + koff);
    v2f b0; b0[0] = co_l[k0 + koff]; b0[1] = co_l[k0 + koff + 1];
    acc0 = __builtin_amdgcn_wmma_f32_16x16x4_f32(false, a0, false, b0,
                                                 (short)0, acc0, false, false);
    v2f a1 = *(const v2f*)(wrow + k0 + 4 + koff);
    v2f b1; b1[0] = co_l[k0 + 4 + koff]; b1[1] = co_l[k0 + 4 + koff + 1];
    acc1 = __builtin_amdgcn_wmma_f32_16x16x4_f32(false, a1, false, b1,
                                                 (short)0, acc1, false, false);
  }
  acc0 += acc1;

  if (lane == 0 || lane == 16) {
    const int half = lane >> 4;                 // 0: rows r0..r0+7, 1: +8..+15
    const float ww = w_write[n];
    #pragma unroll
    for (int i = 0; i < 8; ++i) {
      const int rr = r0 + half * 8 + i;
      const float e = sigmoidf_(acc0[i] + P.we_b[rr]);
      atomicAdd(&er[mbase + half * 8 + i], ww * e);
    }
  }
  __syncthreads();
  if (t < M_) atomicAdd(&erase_acc[t], er[t]);
}

// ---------------- K10: memory update ---------------------------------------
__global__ void k_newmem(InPtrs P, const float* __restrict__ erase_acc,
                         const float* __restrict__ av,
                         const float* __restrict__ w_write,
                         float* __restrict__ dout) {
  const int idx = blockIdx.x * blockDim.x + threadIdx.x;  // 0..131071
  const int n = idx >> 6, m = idx & 63;
  dout[OUT_MEM + idx] =
      P.memory[idx] * (1.0f - erase_acc[m]) + w_write[n] * av[m];
}

// ---------------------------------------------------------------------------
extern "C" void kernel_launch(void* const* d_in, const int* in_sizes, int n_in,
                              void* d_out, int out_size, void* d_ws,
                              size_t ws_size, hipStream_t stream) {
  InPtrs P;
  const float** pp = (const float**)&P;
  for (int i = 0; i < 37; ++i) pp[i] = (const float*)d_in[i];

  float* ws   = (float*)d_ws;
  float* dout = (float*)d_out;

  float* gates = ws + WS_GATES;
  float* co    = ws + WS_CO;
  float* kv    = ws + WS_K;
  float* av    = ws + WS_A;
  float* scal  = ws + WS_SCAL;
  float* slog  = ws + WS_SLOG;
  float* sim   = ws + WS_SIM;
  float* wts   = ws + WS_W;          // [2][2048]
  float* erase = ws + WS_ERASE;
  float* redo  = ws + WS_RED;

  k_lstm_gates<<<512, 256, 0, stream>>>(P, gates);
  k_lstm_cell <<<4,   256, 0, stream>>>(P, gates, co, dout, erase);
  k_small     <<<25,  256, 0, stream>>>(P, co, kv, av, scal);
  k_knorm     <<<2,    32, 0, stream>>>(kv, scal);
  k_slog      <<<512, 256, 0, stream>>>(P, co, slog);
  k_sim       <<<16,  256, 0, stream>>>(P, kv, scal, sim);
  k_address   <<<2,  1024, 0, stream>>>(P, slog, sim, scal, wts, dout);
  k_readout   <<<1,   256, 0, stream>>>(P, wts, redo, dout);
  k_fc        <<<8,   256, 0, stream>>>(P, co, redo, dout);
  k_erase_wmma<<<1024,256, 0, stream>>>(P, co, wts + N_, erase);
  k_newmem    <<<512, 256, 0, stream>>>(P, erase, av, wts + N_, dout);
}